// SlinkyForceODEFunc_14181982011652
// MI455X (gfx1250) — compile-verified
//
#include <hip/hip_runtime.h>

// ---- problem constants ----
#define BATCH 128
#define NPTS  8192
#define NTRI  (NPTS - 2)            // 8190 triplet rows per batch
#define MTOT  (BATCH * NTRI)        // 1,048,320 rows == 16 * 65,520
#define HID   128
#define TILES (MTOT / 16)           // 65,520 16-row tiles
#define WAVES_PER_WG 4
#define NUM_WGS (TILES / WAVES_PER_WG)  // 16,380

// ---- LDS layout ----
// Weights are stored pair-interleaved: sWp[p*STRIDE + n*2 + c] = W[2p+c][n]
// so a B-fragment {W[k][n], W[k+1][n]} is one aligned ds_load_b64.
// Pair-row strides are ≡ 32 (mod 64) dwords so lanes 0-15 (pair p) and
// lanes 16-31 (pair p+1) land in disjoint 32-bank halves (conflict-free).
#define SW12_PS 288   // W1/W2 pair-row stride (dwords), needs >= 256
#define SW3_PS  32    // W3 pair-row stride (dwords), needs >= 32
#define SH_S    130   // h buffer row stride (even -> aligned b64 A-frag loads)
#define SA_S    18    // t/f tile row stride (even)

typedef __attribute__((ext_vector_type(2))) float v2f;
typedef __attribute__((ext_vector_type(8))) float v8f;

__device__ __forceinline__ float swishf(float x) {
    return x / (1.0f + __expf(-x));
}

__device__ __forceinline__ v8f wmma_f32(v2f a, v2f b, v8f c) {
    return __builtin_amdgcn_wmma_f32_16x16x4_f32(
        false, a, false, b, (short)0, c, false, false);
}

__global__ __launch_bounds__(128) void slinky_mlp_wmma(
    const float* __restrict__ y,
    const float* __restrict__ W1, const float* __restrict__ b1,
    const float* __restrict__ W2, const float* __restrict__ b2,
    const float* __restrict__ W3, const float* __restrict__ b3,
    float* __restrict__ out)
{
    __shared__ __align__(16) float sW1p[8  * SW12_PS];   //  9216 B (k-pad to 16)
    __shared__ __align__(16) float sW2p[64 * SW12_PS];   // 73728 B
    __shared__ __align__(16) float sW3p[64 * SW3_PS];    //  8192 B (n-pad to 16)
    __shared__ float sB1[128];
    __shared__ float sB2[128];
    __shared__ float sB3[16];
    __shared__ __align__(16) float sA[WAVES_PER_WG][16 * SA_S];  // t, reused for f
    __shared__ __align__(16) float sH[WAVES_PER_WG][16 * SH_S];  // h1 then h2

    const int tid  = threadIdx.x;          // 0..127
    const int wave = tid >> 5;
    const int lane = tid & 31;
    const int m    = lane & 15;            // tile row for A / column-in-ntile for B,D
    const int hsel = lane >> 4;            // 0: K {0,1}; 1: K {2,3} of each K=4 step

    // ---- cooperative weight staging (pair-interleaved) ----
    for (int idx = tid; idx < 8 * HID; idx += 128) {     // W1: 9x128 padded 16x128
        int p = idx >> 7, n = idx & 127;
        v2f v;
        v.x = (2 * p     < 9) ? W1[(2 * p)     * HID + n] : 0.0f;
        v.y = (2 * p + 1 < 9) ? W1[(2 * p + 1) * HID + n] : 0.0f;
        *(v2f*)&sW1p[p * SW12_PS + n * 2] = v;
    }
    for (int idx = tid; idx < 64 * HID; idx += 128) {    // W2: 128x128
        int p = idx >> 7, n = idx & 127;
        v2f v;
        v.x = W2[(2 * p)     * HID + n];
        v.y = W2[(2 * p + 1) * HID + n];
        *(v2f*)&sW2p[p * SW12_PS + n * 2] = v;
    }
    for (int idx = tid; idx < 64 * 16; idx += 128) {     // W3: 128x9 padded 128x16
        int p = idx >> 4, n = idx & 15;
        v2f v;
        v.x = (n < 9) ? W3[(2 * p)     * 9 + n] : 0.0f;
        v.y = (n < 9) ? W3[(2 * p + 1) * 9 + n] : 0.0f;
        *(v2f*)&sW3p[p * SW3_PS + n * 2] = v;
    }
    if (tid < 128) { sB1[tid] = b1[tid]; sB2[tid] = b2[tid]; }
    if (tid < 16)  { sB3[tid] = (tid < 9) ? b3[tid] : 0.0f; }
    __syncthreads();

    const int  tile  = blockIdx.x * WAVES_PER_WG + wave;
    const long rbase = (long)tile * 16;
    float* myA = &sA[wave][0];
    float* myH = &sH[wave][0];

    // ---- build triplet features t (lanes 0..15, one row each) ----
    if (hsel == 0) {
        long r = rbase + m;
        int  b = (int)(r / NTRI);
        int  i = (int)(r - (long)b * NTRI);
        const float* yb = y + ((long)b * NPTS + i) * 6;
        float xp0 = yb[0],  xp1 = yb[1],  xp2 = yb[2];
        float xm0 = yb[6],  xm1 = yb[7],  xm2 = yb[8];
        float xn0 = yb[12], xn1 = yb[13], xn2 = yb[14];
        float off = atan2f(xm1 - xp1, xm0 - xp0);
        float* a = myA + m * SA_S;
        a[0] = xp2 - off; a[1] = xm2 - off; a[2] = xn2 - off;
        a[3] = xp0; a[4] = xp1; a[5] = xm0; a[6] = xm1; a[7] = xn0; a[8] = xn1;
        #pragma unroll
        for (int c = 9; c < 16; ++c) a[c] = 0.0f;
    }
    __syncthreads();

    // ---- layer 1: h1 = swish(t @ W1 + b1), M=16 K=16(pad) N=128 ----
    {
        v8f acc[8];
        #pragma unroll
        for (int nt = 0; nt < 8; ++nt) acc[nt] = (v8f){0,0,0,0,0,0,0,0};
        #pragma unroll
        for (int k = 0; k < 16; k += 4) {
            v2f af = *(const v2f*)&myA[m * SA_S + k + 2 * hsel];
            int p = (k >> 1) + hsel;
            #pragma unroll
            for (int nt = 0; nt < 8; ++nt) {
                v2f bf = *(const v2f*)&sW1p[p * SW12_PS + (nt * 16 + m) * 2];
                acc[nt] = wmma_f32(af, bf, acc[nt]);
            }
        }
        #pragma unroll
        for (int nt = 0; nt < 8; ++nt) {
            float bv = sB1[nt * 16 + m];
            #pragma unroll
            for (int j = 0; j < 8; ++j) {
                int row = j + 8 * hsel;
                myH[row * SH_S + nt * 16 + m] = swishf(acc[nt][j] + bv);
            }
        }
    }
    __syncthreads();

    // ---- layer 2: h2 = swish(h1 @ W2 + b2), M=16 K=128 N=128 ----
    {
        v8f acc[8];
        #pragma unroll
        for (int nt = 0; nt < 8; ++nt) acc[nt] = (v8f){0,0,0,0,0,0,0,0};
        #pragma unroll 4
        for (int k = 0; k < 128; k += 4) {
            v2f af = *(const v2f*)&myH[m * SH_S + k + 2 * hsel];
            int p = (k >> 1) + hsel;
            #pragma unroll
            for (int nt = 0; nt < 8; ++nt) {
                v2f bf = *(const v2f*)&sW2p[p * SW12_PS + (nt * 16 + m) * 2];
                acc[nt] = wmma_f32(af, bf, acc[nt]);
            }
        }
        __syncthreads();   // all reads of h1 done before overwrite
        #pragma unroll
        for (int nt = 0; nt < 8; ++nt) {
            float bv = sB2[nt * 16 + m];
            #pragma unroll
            for (int j = 0; j < 8; ++j) {
                int row = j + 8 * hsel;
                myH[row * SH_S + nt * 16 + m] = swishf(acc[nt][j] + bv);
            }
        }
    }
    __syncthreads();

    // ---- layer 3: f = h2 @ W3 + b3, M=16 K=128 N=16(pad, 9 valid) ----
    {
        v8f acc = (v8f){0,0,0,0,0,0,0,0};
        #pragma unroll 4
        for (int k = 0; k < 128; k += 4) {
            v2f af = *(const v2f*)&myH[m * SH_S + k + 2 * hsel];
            int p = (k >> 1) + hsel;
            v2f bf = *(const v2f*)&sW3p[p * SW3_PS + m * 2];
            acc = wmma_f32(af, bf, acc);
        }
        __syncthreads();   // t tile fully consumed; reuse sA for f
        float bv = sB3[m];
        #pragma unroll
        for (int j = 0; j < 8; ++j) {
            int row = j + 8 * hsel;
            myA[row * SA_S + m] = acc[j] + bv;
        }
    }
    __syncthreads();

    // ---- scatter-add; positions 0 and N-1 stay zero (pre-memset) ----
    if (hsel == 0) {
        long r = rbase + m;
        int  b = (int)(r / NTRI);
        int  i = (int)(r - (long)b * NTRI);
        const float* f  = myA + m * SA_S;
        float*       ob = out + (long)b * NPTS * 3;
        if (i != 0) {
            atomicAdd(ob + (long)i * 3 + 0, f[3]);
            atomicAdd(ob + (long)i * 3 + 1, f[4]);
            atomicAdd(ob + (long)i * 3 + 2, f[0]);
        }
        atomicAdd(ob + (long)(i + 1) * 3 + 0, f[5]);
        atomicAdd(ob + (long)(i + 1) * 3 + 1, f[6]);
        atomicAdd(ob + (long)(i + 1) * 3 + 2, f[1]);
        if (i != NTRI - 1) {
            atomicAdd(ob + (long)(i + 2) * 3 + 0, f[7]);
            atomicAdd(ob + (long)(i + 2) * 3 + 1, f[8]);
            atomicAdd(ob + (long)(i + 2) * 3 + 2, f[2]);
        }
    }
}

extern "C" void kernel_launch(void* const* d_in, const int* in_sizes, int n_in,
                              void* d_out, int out_size, void* d_ws, size_t ws_size,
                              hipStream_t stream) {
    (void)in_sizes; (void)n_in; (void)d_ws; (void)ws_size; (void)out_size;
    const float* y  = (const float*)d_in[0];
    const float* W1 = (const float*)d_in[1];
    const float* b1 = (const float*)d_in[2];
    const float* W2 = (const float*)d_in[3];
    const float* b2 = (const float*)d_in[4];
    const float* W3 = (const float*)d_in[5];
    const float* b3 = (const float*)d_in[6];
    float* out = (float*)d_out;

    hipMemsetAsync(out, 0, (size_t)BATCH * NPTS * 3 * sizeof(float), stream);
    slinky_mlp_wmma<<<NUM_WGS, 128, 0, stream>>>(y, W1, b1, W2, b2, W3, b3, out);
}